// TasteGNN_78666620994211
// MI455X (gfx1250) — compile-verified
//
#include <hip/hip_runtime.h>
#include <math.h>

// ---------------------------------------------------------------------------
// TasteGNN / HANConv single-edge-type forward for MI455X (gfx1250, wave32).
//   d_out = [ x_ing (copy) | relu(segsum(w * h_src[src])) + x_taste ]
// Simplifications proven from the reference:
//   * beta == softmax(scalar) == 1  ->  W_k, b_k, q unused.
//   * h_dst only feeds a_dst       ->  fold: a_dst = x_taste @ (W_taste@att_dst) + b.att
// Workspace needed: h_src (51.2MB) + a_src + a_dst + v_dst + amax + denom ~= 52.5MB.
// ---------------------------------------------------------------------------

typedef __attribute__((ext_vector_type(2))) float v2f;
typedef __attribute__((ext_vector_type(8))) float v8f;

#define DIM 128
#define NEG_SLOPE 0.2f

// Monotonic order-preserving float <-> u32 encoding (for atomic segment-max).
__device__ __forceinline__ unsigned f32_enc(float f) {
  unsigned u = __float_as_uint(f);
  return (u & 0x80000000u) ? ~u : (u | 0x80000000u);
}
__device__ __forceinline__ float f32_dec(unsigned k) {
  unsigned u = (k & 0x80000000u) ? (k & 0x7FFFFFFFu) : ~k;
  return __uint_as_float(u);
}

__device__ __forceinline__ float lrelu(float x) {
  return x > 0.f ? x : NEG_SLOPE * x;
}

// --- init: zero agg (taste half of d_out), amax=-inf, denom=0 ---------------
__global__ void init_kernel(float* __restrict__ agg, unsigned* __restrict__ amax,
                            float* __restrict__ denom, int n_taste) {
  int i = blockIdx.x * blockDim.x + threadIdx.x;
  if (i < n_taste * DIM) agg[i] = 0.f;
  if (i < n_taste) {
    amax[i]  = f32_enc(-__builtin_inff());
    denom[i] = 0.f;
  }
}

// --- h = X @ W + b using V_WMMA_F32_16X16X4_F32 -----------------------------
// One wave per 16x16 output tile; 8 waves/block cover all 8 N-tiles of D=128;
// W (64KB) staged in LDS once per block (reused by 6250 blocks' worth of rows).
__global__ void __launch_bounds__(256)
proj_wmma(const float* __restrict__ X, const float* __restrict__ W,
          const float* __restrict__ bias, float* __restrict__ H) {
  __shared__ float Wlds[DIM * DIM];
  for (int i = threadIdx.x * 4; i < DIM * DIM; i += 256 * 4)
    *(float4*)(Wlds + i) = *(const float4*)(W + i);
  __syncthreads();

  const int wave = threadIdx.x >> 5;
  const int lane = threadIdx.x & 31;
  const int half = lane >> 4;      // 0: lanes 0-15, 1: lanes 16-31
  const int nl   = lane & 15;
  const int m0   = blockIdx.x * 16;
  const int n0   = wave * 16;

  v8f c;
  const float bv = bias[n0 + nl];  // C/D col = n0+nl for every VGPR row
#pragma unroll
  for (int r = 0; r < 8; r++) c[r] = bv;

  const float* xrow = X + (size_t)(m0 + nl) * DIM;  // A row = m0 + (lane%16)

#pragma unroll
  for (int k = 0; k < DIM; k += 4) {
    const int kk = k + 2 * half;   // this lane's K pair {kk, kk+1}
    v2f a = *(const v2f*)(xrow + kk);                  // A: 16x4 fp32 layout
    v2f b;
    b.x = Wlds[(kk + 0) * DIM + n0 + nl];              // B: 4x16 fp32 layout
    b.y = Wlds[(kk + 1) * DIM + n0 + nl];
#if __has_builtin(__builtin_amdgcn_wmma_f32_16x16x4_f32)
    c = __builtin_amdgcn_wmma_f32_16x16x4_f32(
        /*neg_a=*/false, a, /*neg_b=*/false, b,
        /*c_mod=*/(short)0, c, /*reuse_a=*/false, /*reuse_b=*/false);
#else
    // Mathematically identical scalar fallback (should not be taken).
#pragma unroll
    for (int r = 0; r < 8; r++) {
      const float* xr = X + (size_t)(m0 + r + 8 * half) * DIM;
      float acc = c[r];
      for (int j = 0; j < 4; j++) acc += xr[k + j] * Wlds[(k + j) * DIM + n0 + nl];
      c[r] = acc;
    }
#endif
  }

  // D layout: VGPR r -> row m0 + r + 8*half, col n0 + nl
#pragma unroll
  for (int r = 0; r < 8; r++)
    H[(size_t)(m0 + r + 8 * half) * DIM + n0 + nl] = c[r];
}

// --- fold: v[k] = sum_n W[k,n]*att[n];  c = sum_n b[n]*att[n] ---------------
__global__ void fold_kernel(const float* __restrict__ W, const float* __restrict__ att,
                            const float* __restrict__ b, float* __restrict__ v,
                            float* __restrict__ cscal) {
  const int k = threadIdx.x;  // 128 threads
  float s = 0.f;
  for (int n = 0; n < DIM; n++) s += W[k * DIM + n] * att[n];
  v[k] = s;
  if (k == 0) {
    float t = 0.f;
    for (int n = 0; n < DIM; n++) t += b[n] * att[n];
    *cscal = t;
  }
}

// --- a[i] = H[i,:] . v (+ c) ; one wave32 per row ---------------------------
__global__ void rowdot_kernel(const float* __restrict__ H, const float* __restrict__ v,
                              const float* __restrict__ cscal, float* __restrict__ out,
                              int nrows) {
  const int row  = (blockIdx.x * blockDim.x + threadIdx.x) >> 5;
  const int lane = threadIdx.x & 31;
  if (row >= nrows) return;
  float4 h4 = *(const float4*)(H + (size_t)row * DIM + lane * 4);
  float4 v4 = *(const float4*)(v + lane * 4);
  float s = h4.x * v4.x + h4.y * v4.y + h4.z * v4.z + h4.w * v4.w;
#pragma unroll
  for (int off = 16; off > 0; off >>= 1) s += __shfl_xor(s, off, 32);
  if (lane == 0) out[row] = s + (cscal ? *cscal : 0.f);
}

// --- segment max of leaky_relu(a_src[src]+a_dst[dst]) over dst --------------
__global__ void edge_max_kernel(const int* __restrict__ src, const int* __restrict__ dst,
                                const float* __restrict__ a_src, const float* __restrict__ a_dst,
                                unsigned* __restrict__ amax, int E) {
  int e = blockIdx.x * blockDim.x + threadIdx.x;
  if (e >= E) return;
  float al = lrelu(a_src[src[e]] + a_dst[dst[e]]);
  atomicMax(&amax[dst[e]], f32_enc(al));
}

// --- denom[dst] += exp(alpha - amax[dst]) -----------------------------------
__global__ void edge_denom_kernel(const int* __restrict__ src, const int* __restrict__ dst,
                                  const float* __restrict__ a_src, const float* __restrict__ a_dst,
                                  const unsigned* __restrict__ amax, float* __restrict__ denom,
                                  int E) {
  int e = blockIdx.x * blockDim.x + threadIdx.x;
  if (e >= E) return;
  int d = dst[e];
  float al = lrelu(a_src[src[e]] + a_dst[d]);
  atomicAdd(&denom[d], __expf(al - f32_dec(amax[d])));
}

// --- agg[dst,:] += w_e * h_src[src,:]  (128 threads = one edge) -------------
__global__ void __launch_bounds__(DIM)
edge_scatter_kernel(const int* __restrict__ src, const int* __restrict__ dst,
                    const float* __restrict__ a_src, const float* __restrict__ a_dst,
                    const unsigned* __restrict__ amax, const float* __restrict__ denom,
                    const float* __restrict__ h_src, float* __restrict__ agg) {
  const int e = blockIdx.x;
  __shared__ float wsh;
  const int s = src[e], d = dst[e];
  if (threadIdx.x == 0) {
    float al = lrelu(a_src[s] + a_dst[d]);
    float ex = __expf(al - f32_dec(amax[d]));
    wsh = ex / (denom[d] + 1e-16f);
  }
  __syncthreads();
  const float w = wsh;
  atomicAdd(&agg[(size_t)d * DIM + threadIdx.x],
            h_src[(size_t)s * DIM + threadIdx.x] * w);
}

// --- taste_out = relu(agg) + x_taste  (in place on d_out taste half) --------
__global__ void finalize_kernel(float* __restrict__ agg, const float* __restrict__ x_taste,
                                int n) {
  int i = blockIdx.x * blockDim.x + threadIdx.x;
  if (i >= n) return;
  agg[i] = fmaxf(agg[i], 0.f) + x_taste[i];
}

// ---------------------------------------------------------------------------
extern "C" void kernel_launch(void* const* d_in, const int* in_sizes, int n_in,
                              void* d_out, int out_size, void* d_ws, size_t ws_size,
                              hipStream_t stream) {
  const float* x_ing   = (const float*)d_in[0];
  const float* x_taste = (const float*)d_in[1];
  const int*   src     = (const int*)d_in[2];
  const int*   dst     = (const int*)d_in[3];
  const float* W_ing   = (const float*)d_in[4];
  const float* b_ing   = (const float*)d_in[5];
  const float* W_taste = (const float*)d_in[6];
  const float* b_taste = (const float*)d_in[7];
  const float* att_src = (const float*)d_in[8];
  const float* att_dst = (const float*)d_in[9];
  // d_in[10..12] (W_k, b_k, q): unused — beta == softmax over 1 element == 1.

  const int n_ing   = in_sizes[0] / DIM;   // 100000 (multiple of 16)
  const int n_taste = in_sizes[1] / DIM;   // 50000  (multiple of 16)
  const int E       = in_sizes[2];         // 600000

  // workspace carve-out (256B aligned slots)
  char*  ws  = (char*)d_ws;
  size_t off = 0;
  auto carve = [&](size_t bytes) -> void* {
    void* p = ws + off;
    off = (off + bytes + 255) & ~(size_t)255;
    return p;
  };
  float*    h_src = (float*)   carve((size_t)n_ing * DIM * sizeof(float));
  float*    a_src = (float*)   carve((size_t)n_ing * sizeof(float));
  float*    a_dst = (float*)   carve((size_t)n_taste * sizeof(float));
  float*    v_dst = (float*)   carve(DIM * sizeof(float));
  float*    c_dst = (float*)   carve(256);
  unsigned* amax  = (unsigned*)carve((size_t)n_taste * sizeof(unsigned));
  float*    denom = (float*)   carve((size_t)n_taste * sizeof(float));

  float* out_ing = (float*)d_out;
  float* agg     = out_ing + (size_t)n_ing * DIM;  // taste half doubles as accumulator

  // 1) passthrough ingredient features
  hipMemcpyAsync(out_ing, x_ing, (size_t)n_ing * DIM * sizeof(float),
                 hipMemcpyDeviceToDevice, stream);

  // 2) zero agg, init segment-max/denom
  {
    int n = n_taste * DIM;
    init_kernel<<<(n + 255) / 256, 256, 0, stream>>>(agg, amax, denom, n_taste);
  }

  // 3) h_src = x_ing @ W_ing + b_ing   (fp32 WMMA 16x16x4)
  proj_wmma<<<n_ing / 16, 256, 0, stream>>>(x_ing, W_ing, b_ing, h_src);

  // 4) fold W_taste @ att_dst (kills the h_dst GEMM)
  fold_kernel<<<1, DIM, 0, stream>>>(W_taste, att_dst, b_taste, v_dst, c_dst);

  // 5) a_src = h_src . att_src ;  a_dst = x_taste . v_dst + c
  rowdot_kernel<<<(n_ing + 7) / 8, 256, 0, stream>>>(h_src, att_src, nullptr, a_src, n_ing);
  rowdot_kernel<<<(n_taste + 7) / 8, 256, 0, stream>>>(x_taste, v_dst, c_dst, a_dst, n_taste);

  // 6-7) segment softmax stats over dst
  edge_max_kernel<<<(E + 255) / 256, 256, 0, stream>>>(src, dst, a_src, a_dst, amax, E);
  edge_denom_kernel<<<(E + 255) / 256, 256, 0, stream>>>(src, dst, a_src, a_dst, amax, denom, E);

  // 8) weighted scatter-add of h_src rows into agg
  edge_scatter_kernel<<<E, DIM, 0, stream>>>(src, dst, a_src, a_dst, amax, denom, h_src, agg);

  // 9) relu + residual
  {
    int n = n_taste * DIM;
    finalize_kernel<<<(n + 255) / 256, 256, 0, stream>>>(agg, x_taste, n);
  }
}